// HierarchicalKGNN_37177236914933
// MI455X (gfx1250) — compile-verified
//
#include <hip/hip_runtime.h>

#define DCH 128          // feature dim (both in and out)
#define TS  132          // padded LDS row stride (floats) to avoid bank conflicts
#define BN_EPS 1e-5f

typedef float v2f __attribute__((ext_vector_type(2)));
typedef float v4f __attribute__((ext_vector_type(4)));
typedef float v8f __attribute__((ext_vector_type(8)));

__device__ __forceinline__ void atomAddGlobal(float* p, float v) {
    __hip_atomic_fetch_add(p, v, __ATOMIC_RELAXED, __HIP_MEMORY_SCOPE_AGENT);
}

// ---------------------------------------------------------------- zero scratch
__global__ void zero_kernel(float* __restrict__ aggs, long n4, float* __restrict__ stats) {
    long i = (long)blockIdx.x * blockDim.x + threadIdx.x;
    if (i < n4) {
        v4f z = {0.f, 0.f, 0.f, 0.f};
        *(v4f*)(aggs + i * 4) = z;
    }
    if (blockIdx.x == 0 && threadIdx.x < 256) stats[threadIdx.x] = 0.f;   // colsum+colsumsq
}

// ------------------------------------------------- scatter-add: agg[row] += x[col]
// one wave (32 lanes) per edge; each lane moves 4 consecutive floats (16B)
__global__ void scatter_kernel(const float* __restrict__ x, const int* __restrict__ ei,
                               float* __restrict__ agg, int E) {
    int wave = blockIdx.x * (blockDim.x >> 5) + (threadIdx.x >> 5);
    int lane = threadIdx.x & 31;
    if (wave >= E) return;
    int row = ei[wave];       // ei[0][e]  (segment id)
    int col = ei[E + wave];   // ei[1][e]  (gathered node)
    v4f v = *(const v4f*)(x + (long)col * DCH + lane * 4);
    float* dst = agg + (long)row * DCH + lane * 4;
    atomAddGlobal(dst + 0, v.x);
    atomAddGlobal(dst + 1, v.y);
    atomAddGlobal(dst + 2, v.z);
    atomAddGlobal(dst + 3, v.w);
}

// ---------------- fused triple GEMM (f32 WMMA) + BatchNorm partial statistics
// block = 256 threads = 8 waves; block handles 16 rows x 128 cols of output.
// wave w computes the 16x16 tile at columns [16w, 16w+16).
// out = x@W1^T + aggL@W2l^T + aggG@W2g^T, accumulated in one v8f via
// 96 chained v_wmma_f32_16x16x4_f32.
__global__ void __launch_bounds__(256) gemm_fused_kernel(
        const float* __restrict__ x, const float* __restrict__ aggL,
        const float* __restrict__ aggG,
        const float* __restrict__ W1, const float* __restrict__ W2l,
        const float* __restrict__ W2g,
        float* __restrict__ acc, float* __restrict__ colsum,
        float* __restrict__ colsumsq) {
    __shared__ float tile[3][16][TS];
    __shared__ float csum[DCH];
    __shared__ float csumsq[DCH];

    const int tid = threadIdx.x;
    const long rowBase = (long)blockIdx.x * 16;

    if (tid < DCH) { csum[tid] = 0.f; csumsq[tid] = 0.f; }

    // stage 16x128 tiles of x / aggL / aggG into LDS (float4, coalesced)
    for (int i = tid; i < 3 * 16 * 32; i += 256) {
        int mat = i >> 9;          // 0..2
        int rem = i & 511;
        int r   = rem >> 5;        // 0..15
        int cq  = rem & 31;        // 0..31 (float4 column)
        const float* src = (mat == 0) ? x : (mat == 1) ? aggL : aggG;
        v4f v = *(const v4f*)(src + (rowBase + r) * DCH + cq * 4);
        *(v4f*)&tile[mat][r][cq * 4] = v;
    }
    __syncthreads();

    const int wave = tid >> 5;
    const int lane = tid & 31;
    const int m16  = lane & 15;        // row within A / column within B,C,D
    const int hi   = lane >> 4;        // K sub-pair select (A/B), M+8 select (C/D)
    const int col  = wave * 16 + m16;  // output column this lane owns

    v8f c = {0.f, 0.f, 0.f, 0.f, 0.f, 0.f, 0.f, 0.f};

    #pragma unroll
    for (int mat = 0; mat < 3; ++mat) {
        const float* tb = &tile[mat][m16][2 * hi];                 // A: LDS b64 loads
        const float* W  = (mat == 0) ? W1 : (mat == 1) ? W2l : W2g;
        const float* wb = W + (long)col * DCH + 2 * hi;            // B: L2-hit b64 loads
        #pragma unroll 8
        for (int k = 0; k < DCH; k += 4) {
            v2f a = *(const v2f*)(tb + k);   // A[m16][k+2hi .. k+2hi+1]
            v2f b = *(const v2f*)(wb + k);   // W[col][k+2hi .. k+2hi+1]
            c = __builtin_amdgcn_wmma_f32_16x16x4_f32(
                    false, a, false, b, (short)0, c, false, false);
        }
    }

    // D layout: vgpr r -> M = r + 8*hi, N = m16.  Store + BN partials.
    float s = 0.f, s2 = 0.f;
    #pragma unroll
    for (int r = 0; r < 8; ++r) {
        float v = c[r];
        acc[(rowBase + r + 8 * hi) * DCH + col] = v;
        s  += v;
        s2 += v * v;
    }
    __hip_atomic_fetch_add(&csum[col],   s,  __ATOMIC_RELAXED, __HIP_MEMORY_SCOPE_WORKGROUP);
    __hip_atomic_fetch_add(&csumsq[col], s2, __ATOMIC_RELAXED, __HIP_MEMORY_SCOPE_WORKGROUP);
    __syncthreads();

    if (tid < DCH) {
        atomAddGlobal(&colsum[tid],   csum[tid]);
        atomAddGlobal(&colsumsq[tid], csumsq[tid]);
    }
}

// --------------------------------------- fold BN stats into per-column scale/shift
__global__ void bn_finalize_kernel(const float* __restrict__ colsum,
                                   const float* __restrict__ colsumsq,
                                   const float* __restrict__ gamma,
                                   const float* __restrict__ beta,
                                   float* __restrict__ ss, float invN) {
    int cidx = threadIdx.x;
    if (cidx >= DCH) return;
    float mean = colsum[cidx] * invN;
    float var  = colsumsq[cidx] * invN - mean * mean;
    float sc   = gamma[cidx] * rsqrtf(var + BN_EPS);
    ss[cidx]        = sc;
    ss[DCH + cidx]  = beta[cidx] - mean * sc;
}

// ------------------------------------------------------- normalize + ReLU epilogue
__global__ void norm_relu_kernel(const float* __restrict__ acc,
                                 const float* __restrict__ ss,
                                 float* __restrict__ out, long total4) {
    long i = (long)blockIdx.x * blockDim.x + threadIdx.x;
    if (i >= total4) return;
    v4f v = *(const v4f*)(acc + i * 4);
    int c0 = (int)((i * 4) & (DCH - 1));
    v4f r;
    #pragma unroll
    for (int j = 0; j < 4; ++j)
        r[j] = fmaxf(v[j] * ss[c0 + j] + ss[DCH + c0 + j], 0.f);
    *(v4f*)(out + i * 4) = r;
}

// =============================================================================
extern "C" void kernel_launch(void* const* d_in, const int* in_sizes, int n_in,
                              void* d_out, int out_size, void* d_ws, size_t ws_size,
                              hipStream_t stream) {
    const float* x     = (const float*)d_in[0];
    const int*   eL    = (const int*)  d_in[1];
    const int*   eG    = (const int*)  d_in[2];
    const float* W1    = (const float*)d_in[3];
    const float* W2l   = (const float*)d_in[4];
    const float* W2g   = (const float*)d_in[5];
    const float* gamma = (const float*)d_in[6];
    const float* beta  = (const float*)d_in[7];
    float* out = (float*)d_out;

    const int N = in_sizes[0] / DCH;   // 100000
    const int E = in_sizes[1] / 2;     // 1000000

    float* ws   = (float*)d_ws;
    size_t npd  = (size_t)N * DCH;
    float* aggL = ws;
    float* aggG = ws + npd;
    float* acc  = ws + 2 * npd;
    float* stats = ws + 3 * npd;       // [colsum(128) | colsumsq(128) | scale(128) | shift(128)]
    float* colsum   = stats;
    float* colsumsq = stats + DCH;
    float* ss       = stats + 2 * DCH;

    // 1) zero aggregation buffers + stats
    long zero4 = (long)(2 * npd) / 4;
    zero_kernel<<<(int)((zero4 + 255) / 256), 256, 0, stream>>>(ws, zero4, stats);

    // 2) edge scatter-adds (one wave per edge)
    int sblocks = (E + 7) / 8;
    scatter_kernel<<<sblocks, 256, 0, stream>>>(x, eL, aggL, E);
    scatter_kernel<<<sblocks, 256, 0, stream>>>(x, eG, aggG, E);

    // 3) fused triple WMMA GEMM + BN partial sums
    gemm_fused_kernel<<<N / 16, 256, 0, stream>>>(x, aggL, aggG, W1, W2l, W2g,
                                                  acc, colsum, colsumsq);

    // 4) BN stat finalize
    bn_finalize_kernel<<<1, DCH, 0, stream>>>(colsum, colsumsq, gamma, beta, ss,
                                              1.0f / (float)N);

    // 5) normalize + ReLU
    long total4 = (long)npd / 4;
    norm_relu_kernel<<<(int)((total4 + 255) / 256), 256, 0, stream>>>(acc, ss, out, total4);
}